// BioConvolution_36653250904141
// MI455X (gfx1250) — compile-verified
//
#include <hip/hip_runtime.h>

typedef float v2f __attribute__((ext_vector_type(2)));
typedef float v8f __attribute__((ext_vector_type(8)));
typedef int   v4i __attribute__((vector_size(4 * sizeof(int))));

// Problem constants: X(64,64,64,64) f32, filters(1,256,4,4,64,128) f32,
// bias(128) f32, out(64,16,16,128) f32. Per-location GEMM: M=64, K=1024, N=128.
#define NIMG      64
#define KTOT      1024
#define NFOUT     128
#define KC        32            // K-chunk per pipeline stage
#define NCHUNK    (KTOT / KC)   // 32
#define SA_STRIDE 36            // 32 + 4 pad  (conflict-free column reads)
#define SB_STRIDE 136           // 128 + 8 pad (half-wave rows -> +16 banks)
#define SA_ELEMS  (NIMG * SA_STRIDE)   // 2304 floats (~9 KB)
#define SB_ELEMS  (KC * SB_STRIDE)     // 4352 floats (~17 KB)

// Typed address-space pointers for the async-copy builtin:
//   __builtin_amdgcn_global_load_async_to_lds_b128(AS1 v4i*, AS3 v4i*, int, int)
// Generic pointers on AMDGPU carry the LDS byte offset in the low 32 bits
// (aperture scheme) and the full 64-bit address for global, so integer
// round-trips are exact.
typedef __attribute__((address_space(1))) v4i gv4i;
typedef __attribute__((address_space(3))) v4i lv4i;
static __device__ __forceinline__ gv4i* as_global(const void* p) {
  return (gv4i*)(unsigned long long)p;
}
static __device__ __forceinline__ lv4i* as_lds(void* p) {
  return (lv4i*)(unsigned int)(unsigned long long)p;
}

__global__ __launch_bounds__(256)
void bioconv_wmma_f32(const float* __restrict__ X,
                      const float* __restrict__ F,
                      const float* __restrict__ bias,
                      float* __restrict__ out) {
  __shared__ float sA[2][SA_ELEMS];   // 64 images x 32 K, double-buffered
  __shared__ float sB[2][SB_ELEMS];   // 32 K x 128 FOUT, double-buffered

  const int l    = blockIdx.x;      // location 0..255
  const int r    = l >> 4;          // patch row (rows = 16)
  const int c    = l & 15;          // patch col (cols = 16)
  const int t    = threadIdx.x;     // 0..255 (8 wave32)
  const int wave = t >> 5;          // 0..7  -> FOUT tile
  const int lane = t & 31;
  const int half = lane >> 4;       // 0: lanes 0-15, 1: lanes 16-31
  const int mrow = lane & 15;       // M (or N) index within tile

  const float* Fl = F + (size_t)l * KTOT * NFOUT;

  // Issue the async global->LDS copies for chunk `kc` into buffer `buf`.
  auto stage = [&](int kc, int buf) {
    // A chunk: 64 images x 32 K. K = kc*32..+31 is one contiguous 32-float
    // run in X: row r*4 + kc/8, inner offset (kc%8)*32. 8 floats/thread.
    {
      const int n   = t >> 2;                  // image 0..63
      const int q   = t & 3;                   // quarter of 32 floats
      const int i   = kc >> 3;                 // fh index
      const int off = (kc & 7) * 32 + q * 8;
      const float* g = X + (size_t)n * 262144 + (size_t)(r * 4 + i) * 4096
                         + c * 256 + off;
      float* d = &sA[buf][n * SA_STRIDE + q * 8];
      __builtin_amdgcn_global_load_async_to_lds_b128(as_global(g),     as_lds(d),     0, 0);
      __builtin_amdgcn_global_load_async_to_lds_b128(as_global(g + 4), as_lds(d + 4), 0, 0);
    }
    // B chunk: 32 K x 128 FOUT, contiguous 4096 floats. 16 floats/thread.
    {
      const float* g = Fl + (size_t)kc * KC * NFOUT + t * 16;
      const int k  = t >> 3;
      const int f0 = (t & 7) * 16;
      float* d = &sB[buf][k * SB_STRIDE + f0];
#pragma unroll
      for (int u = 0; u < 4; ++u)
        __builtin_amdgcn_global_load_async_to_lds_b128(
            as_global(g + u * 4), as_lds(d + u * 4), 0, 0);
    }
  };

  // 4 M-tiles of 16x16 f32 accumulators per wave
  v8f acc[4];
#pragma unroll
  for (int m = 0; m < 4; ++m)
#pragma unroll
    for (int g = 0; g < 8; ++g) acc[m][g] = 0.0f;

  stage(0, 0);   // prologue: fill buffer 0

  for (int kc = 0; kc < NCHUNK; ++kc) {
    const int cur = kc & 1;

    // Drain this wave's async copies (chunk kc), then make them visible to
    // all waves. The barrier also proves every wave finished computing on
    // the *other* buffer, so it is safe to overwrite next.
    asm volatile("s_wait_asynccnt 0x0" ::: "memory");
    __syncthreads();

    if (kc + 1 < NCHUNK) stage(kc + 1, cur ^ 1);   // overlap with compute

    // ---- Compute: 8 K-steps of V_WMMA_F32_16X16X4_F32 per M-tile.
    const int col = wave * 16 + mrow;   // N index this lane owns in B/C/D
#pragma unroll
    for (int kk = 0; kk < KC / 4; ++kk) {
      const int kb = kk * 4 + half * 2;  // this lane's K pair
      // B fragment (4x16 K-major): VGPR0 = rows K0/K2, VGPR1 = rows K1/K3
      v2f b;
      b.x = sB[cur][(kb + 0) * SB_STRIDE + col];
      b.y = sB[cur][(kb + 1) * SB_STRIDE + col];
#pragma unroll
      for (int m = 0; m < 4; ++m) {
        // A fragment (16x4): lane = M row; K pair contiguous in sA row
        const float* ap = &sA[cur][(m * 16 + mrow) * SA_STRIDE + kb];
        v2f a; a.x = ap[0]; a.y = ap[1];
        acc[m] = __builtin_amdgcn_wmma_f32_16x16x4_f32(
            /*neg_a=*/false, a, /*neg_b=*/false, b,
            /*c_mod=*/(short)0, acc[m], /*reuse_a=*/false, /*reuse_b=*/false);
      }
    }
  }

  // ---- Epilogue: bias + ReLU + store. D layout: VGPR g holds M = half*8+g,
  // lane%16 = N. out[n, r, c, f] strides (32768, 2048, 128, 1).
  const int f   = wave * 16 + mrow;
  const float bv = bias[f];
  float* ob = out + (size_t)r * 2048 + (size_t)c * 128 + f;
#pragma unroll
  for (int m = 0; m < 4; ++m) {
#pragma unroll
    for (int g = 0; g < 8; ++g) {
      const int n = m * 16 + half * 8 + g;
      float v = acc[m][g] + bv;
      ob[(size_t)n * 32768] = v > 0.0f ? v : 0.0f;
    }
  }
}

extern "C" void kernel_launch(void* const* d_in, const int* in_sizes, int n_in,
                              void* d_out, int out_size, void* d_ws, size_t ws_size,
                              hipStream_t stream) {
  const float* X    = (const float*)d_in[0];   // 64*64*64*64
  const float* Flt  = (const float*)d_in[1];   // 256*4*4*64*128
  const float* bias = (const float*)d_in[2];   // 128
  float* out        = (float*)d_out;           // 64*16*16*128
  (void)in_sizes; (void)n_in; (void)out_size; (void)d_ws; (void)ws_size;

  bioconv_wmma_f32<<<256, 256, 0, stream>>>(X, Flt, bias, out);
}